// SimVQ_64278480552679
// MI455X (gfx1250) — compile-verified
//
#include <hip/hip_runtime.h>
#include <hip/hip_bf16.h>
#include <math.h>

#define DIM   512
#define CDIM  128
#define CSIZE 8192
#define ROWS  16384      // 4 * 4096
#define EPSV  1e-6f

// padded LDS row stride (in floats): 512 + 4 (TDM pad: 2 dwords per 256 dwords)
#define LSTR  516

typedef __attribute__((ext_vector_type(2))) float v2f;
typedef __attribute__((ext_vector_type(8))) float v8f;
typedef __attribute__((ext_vector_type(4))) unsigned int v4u;
typedef __attribute__((ext_vector_type(8))) int v8i;
typedef __attribute__((ext_vector_type(4))) int v4i;

__device__ __forceinline__ v8f wmma_f32x4(v2f a, v2f b, v8f c) {
  // D = A(16x4 f32) * B(4x16 f32) + C(16x16 f32)
  return __builtin_amdgcn_wmma_f32_16x16x4_f32(false, a, false, b, (short)0, c,
                                               false, false);
}

__device__ __forceinline__ float selu_f(float v) {
  return 1.0507009873554805f *
         (v > 0.f ? v : 1.6732632423543772f * (__expf(v) - 1.f));
}

// ---------------------------------------------------------------------------
// TDM: DMA a [tile_rows x 512] f32 tile (row stride 512 elems in global) into
// LDS at lds_addr, inserting 2 pad dwords every 256 dwords -> LDS row stride
// 516 floats. D# per CDNA5 ISA (group0/group1 bitfields).
// 6-arg builtin on this toolchain: (g0 v4u, g1 v8i, g2 v4i, g3 v4i, ext v8i,
// cpol i32); groups 2/3/ext unused for a 2D tile -> zeros.
// ---------------------------------------------------------------------------
__device__ __forceinline__ void tdm_load_tile(unsigned int lds_addr,
                                              const float* gptr, int tile_rows) {
  unsigned long long ga = (unsigned long long)(uintptr_t)gptr;
  v4u g0 = { 1u,                                   // [1:0] count=1, rest 0
             lds_addr,                             // [63:32] lds_addr (bytes)
             (unsigned int)ga,                     // [95:64] global_addr lo
             (unsigned int)(ga >> 32) | (2u << 30) };  // [120:96] addr hi | type=2
  v8i g1 = { (int)((2u << 16)      // data_size = 4 bytes
                 | (1u << 20)      // pad_enable
                 | (7u << 22)      // pad_interval: 256 dwords
                 | (1u << 25)),    // pad_amount: 2 dwords
             (int)(512u << 16),    // [47:32] abar=0 | [63:48] tensor_dim0 lo16 = 512
             (int)(8192u << 16),   // [79:64] tensor_dim0 hi = 0 | [95:80] tensor_dim1 lo16
             (int)(512u << 16),    // [111:96] tensor_dim1 hi = 0 | [127:112] tile_dim0
             tile_rows,            // [143:128] tile_dim1 | [159:144] tile_dim2 = 0
             512,                  // [207:160] tensor_dim0_stride = 512 (lo32)
             0,                    // stride hi | tensor_dim1_stride lo (unused, 2D)
             0 };
  v4i z4 = { 0, 0, 0, 0 };
  v8i z8 = { 0, 0, 0, 0, 0, 0, 0, 0 };
  __builtin_amdgcn_tensor_load_to_lds(g0, g1, z4, z4, z8, 0);
}

// ---------------------------------------------------------------------------
// h = codebook @ W1 + b1              [8192, 512], K = 128
// ---------------------------------------------------------------------------
__global__ void k_gemm_h(const float* __restrict__ cb, const float* __restrict__ W1,
                         const float* __restrict__ b1, float* __restrict__ h) {
  const int lane = threadIdx.x & 31;
  const int warp = threadIdx.x >> 5;
  const int wid  = blockIdx.x * 8 + warp;      // 512*32 = 16384 tiles
  const int tm = wid >> 5;
  const int tn = wid & 31;
  const int row0 = tm * 16, col0 = tn * 16;
  const int half = lane >> 4, l = lane & 15;

  v8f acc = {};
  for (int k = 0; k < CDIM; k += 4) {
    v2f a = *(const v2f*)&cb[(row0 + l) * CDIM + k + 2 * half];
    v2f b;
    b.x = W1[(k + 2 * half) * DIM + col0 + l];
    b.y = W1[(k + 2 * half + 1) * DIM + col0 + l];
    acc = wmma_f32x4(a, b, acc);
  }
  const float bb = b1[col0 + l];
#pragma unroll
  for (int i = 0; i < 8; ++i)
    h[(row0 + i + 8 * half) * DIM + col0 + l] = acc[i] + bb;
}

// ---------------------------------------------------------------------------
// icb = h + selu(h) @ W2 + b2         [8192, 512], K = 512
// ---------------------------------------------------------------------------
__global__ void k_gemm_icb(const float* __restrict__ h, const float* __restrict__ W2,
                           const float* __restrict__ b2, float* __restrict__ icb) {
  const int lane = threadIdx.x & 31;
  const int warp = threadIdx.x >> 5;
  const int wid  = blockIdx.x * 8 + warp;
  const int tm = wid >> 5;
  const int tn = wid & 31;
  const int row0 = tm * 16, col0 = tn * 16;
  const int half = lane >> 4, l = lane & 15;

  v8f acc = {};
  for (int k = 0; k < DIM; k += 4) {
    v2f hv = *(const v2f*)&h[(row0 + l) * DIM + k + 2 * half];
    v2f a;
    a.x = selu_f(hv.x);
    a.y = selu_f(hv.y);
    v2f b;
    b.x = W2[(k + 2 * half) * DIM + col0 + l];
    b.y = W2[(k + 2 * half + 1) * DIM + col0 + l];
    acc = wmma_f32x4(a, b, acc);
  }
  const float bb = b2[col0 + l];
#pragma unroll
  for (int i = 0; i < 8; ++i) {
    const int m = row0 + i + 8 * half, n = col0 + l;
    icb[m * DIM + n] = h[m * DIM + n] + bb + acc[i];
  }
}

// ---------------------------------------------------------------------------
// n2[c] = sum_d icb[c,d]^2
// ---------------------------------------------------------------------------
__global__ void k_norm2(const float* __restrict__ icb, float* __restrict__ n2) {
  __shared__ float red[128];
  const int r = blockIdx.x, t = threadIdx.x;
  float s = 0.f;
  for (int c = t; c < DIM; c += 128) {
    const float v = icb[r * DIM + c];
    s += v * v;
  }
  red[t] = s;
  __syncthreads();
  for (int off = 64; off > 0; off >>= 1) {
    if (t < off) red[t] += red[t + off];
    __syncthreads();
  }
  if (t == 0) n2[r] = red[0];
}

// ---------------------------------------------------------------------------
// Fused distance GEMM + argmin. Block = 256 threads = 8 waves = 128 x-rows.
// x block (128x512) and icb tile (16x512) both live in LDS (516-float padded
// rows via TDM padding -> conflict-free ds_load_b64 fragments).
// d = ||icb_c||^2 - 2 * x.icb_c   (||x||^2 constant per row -> dropped)
// ---------------------------------------------------------------------------
__global__ void k_search(const float* __restrict__ x, const float* __restrict__ icb,
                         const float* __restrict__ n2,
                         float* __restrict__ out_idx_f, int* __restrict__ out_idx_i) {
  __shared__ float xs[128 * LSTR];   // 258 KB
  __shared__ float bt[16 * LSTR];    //  33 KB   (total ~291 KB < 320 KB/WGP)

  const int lane  = threadIdx.x & 31;
  const int warp  = threadIdx.x >> 5;
  const int half  = lane >> 4, l = lane & 15;
  const int mrow0 = blockIdx.x * 128 + warp * 16;

  // stage x block once (TDM, wave 0 issues; EXEC-independent wave-level DMA)
  if (warp == 0)
    tdm_load_tile((unsigned int)(uintptr_t)&xs[0], &x[(size_t)blockIdx.x * 128 * DIM], 128);

  float bestv[8];
  int   besti[8];
#pragma unroll
  for (int i = 0; i < 8; ++i) { bestv[i] = 3.4e38f; besti[i] = 0; }

  const float* arow = &xs[(warp * 16 + l) * LSTR + 2 * half];

  for (int nt = 0; nt < CSIZE / 16; ++nt) {
    __syncthreads();                       // previous tile fully consumed
    if (warp == 0) {
      tdm_load_tile((unsigned int)(uintptr_t)&bt[0], &icb[(size_t)nt * 16 * DIM], 16);
      __builtin_amdgcn_s_wait_tensorcnt(0);   // covers xs DMA too on nt==0
    }
    __syncthreads();

    v8f acc = {};
#pragma unroll 4
    for (int k = 0; k < DIM; k += 4) {
      const int pk = k + 2 * ((k + 2) >> 8);   // skip TDM pad at dword 256
      v2f a = *(const v2f*)(arow + pk);
      v2f b = *(const v2f*)&bt[l * LSTR + pk + 2 * half];
      acc = wmma_f32x4(a, b, acc);
    }

    const int   c  = nt * 16 + l;
    const float nn = n2[c];
#pragma unroll
    for (int i = 0; i < 8; ++i) {
      const float d = nn - 2.f * acc[i];
      if (d < bestv[i]) { bestv[i] = d; besti[i] = c; }  // strict < keeps lowest c
    }
  }

  // argmin across the 16 lanes of each half (rows i / 8+i of the wave's tile)
#pragma unroll
  for (int i = 0; i < 8; ++i) {
    float bv = bestv[i];
    int   bi = besti[i];
#pragma unroll
    for (int m = 8; m >= 1; m >>= 1) {
      const float ov = __shfl_xor(bv, m, 32);
      const int   oi = __shfl_xor(bi, m, 32);
      if (ov < bv || (ov == bv && oi < bi)) { bv = ov; bi = oi; }
    }
    if (l == 0) {
      const int row = mrow0 + i + 8 * half;
      out_idx_f[row] = (float)bi;
      out_idx_i[row] = bi;
    }
  }
}

// ---------------------------------------------------------------------------
// Gather + Householder rotation + commit-loss partial (f64 accumulation).
// out = scale*(ax*x + aq*q) from {Σx², Σq², Σxq}.
// ---------------------------------------------------------------------------
__global__ void k_rotate(const float* __restrict__ x, const float* __restrict__ icb,
                         const int* __restrict__ idx, float* __restrict__ outq,
                         double* __restrict__ loss_acc) {
  __shared__ float sxx[256], sqq[256], sxq[256];
  __shared__ float coef[2];
  const int r = blockIdx.x, t = threadIdx.x;
  const int c = idx[r];
  const float* xr = &x[r * DIM];
  const float* qr = &icb[c * DIM];

  const float x0 = xr[t], x1 = xr[t + 256];
  const float q0 = qr[t], q1 = qr[t + 256];
  sxx[t] = x0 * x0 + x1 * x1;
  sqq[t] = q0 * q0 + q1 * q1;
  sxq[t] = x0 * q0 + x1 * q1;
  __syncthreads();
  for (int off = 128; off > 0; off >>= 1) {
    if (t < off) {
      sxx[t] += sxx[t + off];
      sqq[t] += sqq[t + off];
      sxq[t] += sxq[t + off];
    }
    __syncthreads();
  }
  if (t == 0) {
    const float ns2 = sxx[0], nt2 = sqq[0], xq = sxq[0];
    const float ns  = sqrtf(ns2), ntg = sqrtf(nt2);
    const float nsm = fmaxf(ns, EPSV), ntm = fmaxf(ntg, EPSV);
    const float su  = ns2 / nsm;                               // x . u
    const float nw2 = ns2 / (nsm * nsm) + 2.f * xq / (nsm * ntm) + nt2 / (ntm * ntm);
    const float nwm = fmaxf(sqrtf(nw2), EPSV);
    const float swu = ns2 / nsm + xq / ntm;                    // x . (u + q_hat)
    const float cw  = 2.f * (swu / nwm) / nwm;
    const float ax  = 1.f - cw / nsm;
    const float aq  = (2.f * su - cw) / ntm;
    const float scale = ntg / nsm;
    coef[0] = ax * scale;
    coef[1] = aq * scale;
    atomicAdd(loss_acc, (double)(ns2 - 2.f * xq + nt2));       // Σ(x-q)², f64
  }
  __syncthreads();
  const float ax = coef[0], aq = coef[1];
  outq[r * DIM + t]       = ax * x0 + aq * q0;
  outq[r * DIM + t + 256] = ax * x1 + aq * q1;
}

__global__ void k_zero(double* __restrict__ p) { *p = 0.0; }

__global__ void k_final(const double* __restrict__ acc, float* __restrict__ out) {
  // commit = mean((sg(x)-q)^2) + 0.25*mean((x-sg(q))^2) = 1.25*mean((x-q)^2)
  *out = (float)(1.25 * (*acc) / (double)(ROWS * DIM));
}

// ---------------------------------------------------------------------------
extern "C" void kernel_launch(void* const* d_in, const int* in_sizes, int n_in,
                              void* d_out, int out_size, void* d_ws, size_t ws_size,
                              hipStream_t stream) {
  const float* x  = (const float*)d_in[0];   // [4,4096,512]
  const float* cb = (const float*)d_in[1];   // [8192,128]
  const float* W1 = (const float*)d_in[2];   // [128,512]
  const float* b1 = (const float*)d_in[3];   // [512]
  const float* W2 = (const float*)d_in[4];   // [512,512]
  const float* b2 = (const float*)d_in[5];   // [512]

  float* out_q    = (float*)d_out;                      // 8,388,608
  float* out_idx  = out_q + (size_t)ROWS * DIM;         // 16,384 (as float)
  float* out_loss = out_idx + ROWS;                     // 1

  float*  ws   = (float*)d_ws;
  float*  wh   = ws;                                    // 8192*512
  float*  wicb = wh + (size_t)CSIZE * DIM;              // 8192*512
  float*  wn2  = wicb + (size_t)CSIZE * DIM;            // 8192
  int*    widx = (int*)(wn2 + CSIZE);                   // 16384
  double* wacc = (double*)(widx + ROWS);                // 1 (8B, 8-aligned: offset even)

  k_zero<<<1, 1, 0, stream>>>(wacc);
  k_gemm_h<<<(CSIZE / 16) * (DIM / 16) / 8, 256, 0, stream>>>(cb, W1, b1, wh);
  k_gemm_icb<<<(CSIZE / 16) * (DIM / 16) / 8, 256, 0, stream>>>(wh, W2, b2, wicb);
  k_norm2<<<CSIZE, 128, 0, stream>>>(wicb, wn2);
  k_search<<<ROWS / 128, 256, 0, stream>>>(x, wicb, wn2, out_idx, widx);
  k_rotate<<<ROWS, 256, 0, stream>>>(x, wicb, widx, out_q, wacc);
  k_final<<<1, 1, 0, stream>>>(wacc, out_loss);
}